// GigaMesher9000_12687333392689
// MI455X (gfx1250) — compile-verified
//
#include <hip/hip_runtime.h>

// MI455X / gfx1250 implementation.
// Two-pass: pass1 computes the two 3->10->10 branches per 16-row tile with
// V_WMMA_F32_16X16X4_F32 and atomically accumulates the global sum-of-squares;
// pass2 recomputes s,p, applies the rsqrt coupling, and runs 10->100->3 fully
// on the WMMA pipe, using per-wave LDS scratch for C-layout -> A-layout
// conversion (bank-conflict-free strides 17 / 113).
//
// All in-loop guarded loads are branch-free: zero-padded tiles make the LDS
// A-fragment reads unconditional, and row masking uses clamped indices +
// v_cndmask on a wave-uniform full/partial-tile split (EXEC stays all-1s for
// every WMMA).

typedef float v2f __attribute__((ext_vector_type(2)));
typedef float v8f __attribute__((ext_vector_type(8)));

#define TILE_STRIDE 17   // 16x16 conversion tile; 17 mod 64 coprime -> no bank conflicts
#define H_STRIDE    113  // 16x112 hidden tile;   113 mod 64 = 49, coprime -> no conflicts

__device__ __forceinline__ v8f wmma4(v2f a, v2f b, v8f c) {
  // D = A(16x4,f32) x B(4x16,f32) + C(16x16,f32)
  return __builtin_amdgcn_wmma_f32_16x16x4_f32(
      /*neg_a=*/false, a, /*neg_b=*/false, b,
      /*c_mod=*/(short)0, c, /*reuse_a=*/false, /*reuse_b=*/false);
}

__device__ __forceinline__ v8f splat8(float v) {
  v8f r;
#pragma unroll
  for (int i = 0; i < 8; ++i) r[i] = v;
  return r;
}

__device__ __forceinline__ v8f relu8(v8f a) {
#pragma unroll
  for (int i = 0; i < 8; ++i) a[i] = fmaxf(a[i], 0.f);
  return a;
}

// B fragment (4x16 K-major slice of row-major KxN weight), zero-padded.
// Lane l holds N = l%16, K = kbase + 2*(l/16) + {0,1}.
// Unconditional clamped loads + VALU select (init-time only, but keeps the
// prologue free of exec-mask churn too).
__device__ __forceinline__ v2f load_bfrag(const float* __restrict__ W, int ldw,
                                          int kbase, int nbase, int Kv, int Nv,
                                          int lane) {
  int n  = nbase + (lane & 15);
  int k  = kbase + 2 * (lane >> 4);
  int nc = n < Nv ? n : (Nv - 1);
  int k0 = k < Kv ? k : (Kv - 1);
  int k1 = (k + 1) < Kv ? (k + 1) : (Kv - 1);
  float t0 = W[k0 * ldw + nc];
  float t1 = W[k1 * ldw + nc];
  v2f f;
  f.x = (k     < Kv && n < Nv) ? t0 : 0.f;
  f.y = (k + 1 < Kv && n < Nv) ? t1 : 0.f;
  return f;
}

// Per-lane bias value for C-layout broadcast (every C VGPR of lane l is column l%16).
__device__ __forceinline__ float load_biaslane(const float* __restrict__ b,
                                               int nbase, int Nv, int lane) {
  int n  = nbase + (lane & 15);
  int nc = n < Nv ? n : (Nv - 1);
  float t = b[nc];
  return (n < Nv) ? t : 0.f;
}

// Store a 16x16 f32 C/D tile to LDS (row m = v + 8*(lane/16), col n = lane%16).
__device__ __forceinline__ void store_ctile(float* t, v8f c, int lane) {
  int n  = lane & 15;
  int mb = (lane >> 4) * 8;
#pragma unroll
  for (int v = 0; v < 8; ++v) t[(mb + v) * TILE_STRIDE + n] = c[v];
}

// Load an A fragment (16x4) from an LDS tile: lane l holds M = l%16,
// K = kbase + 2*(l/16) + {0,1}. Unconditional: padding columns hold exact 0s.
__device__ __forceinline__ v2f load_afrag(const float* t, int stride, int kbase,
                                          int lane) {
  int m = lane & 15;
  int k = kbase + 2 * (lane >> 4);
  const float* p = t + m * stride + k;
  v2f a;
  a.x = p[0];
  a.y = p[1];
  return a;
}

struct SPFrags {
  v2f sw1f, pw1f;
  v2f sw2f[3], pw2f[3];
  float sb1v, sb2v, pb1v, pb2v;
};

__device__ __forceinline__ void load_sp_frags(
    SPFrags& F, const float* sw1, const float* sb1, const float* sw2,
    const float* sb2, const float* pw1, const float* pb1, const float* pw2,
    const float* pb2, int lane) {
  F.sw1f = load_bfrag(sw1, 10, 0, 0, 3, 10, lane);
  F.pw1f = load_bfrag(pw1, 10, 0, 0, 3, 10, lane);
#pragma unroll
  for (int kk = 0; kk < 3; ++kk) {
    F.sw2f[kk] = load_bfrag(sw2, 10, 4 * kk, 0, 10, 10, lane);
    F.pw2f[kk] = load_bfrag(pw2, 10, 4 * kk, 0, 10, 10, lane);
  }
  F.sb1v = load_biaslane(sb1, 0, 10, lane);
  F.sb2v = load_biaslane(sb2, 0, 10, lane);
  F.pb1v = load_biaslane(pb1, 0, 10, lane);
  F.pb2v = load_biaslane(pb2, 0, 10, lane);
}

// s = relu(x[:, :3]@sw1 + sb1)@sw2 + sb2  (and same for p with cols 3:6),
// for one 16-row tile; results in WMMA C layout (cols >= 10 are exactly 0).
// FULL=true: all 16 rows valid, no masking. FULL=false: clamp + cndmask.
template <bool FULL>
__device__ __forceinline__ void compute_sp(const float* __restrict__ x, long r0,
                                           int rows, const SPFrags& F,
                                           float* tile, int lane, int c0,
                                           int c1, bool lanelow, v8f& sOut,
                                           v8f& pOut) {
  int m = lane & 15;
  int rm = FULL ? m : (m < rows ? m : 0);
  const float* row = x + (r0 + rm) * 6;
  float t0 = row[c0];
  float t1 = row[c1];
  float t2 = row[3 + c0];
  float t3 = row[3 + c1];
  if (!FULL) {
    bool ok = (m < rows);
    t0 = ok ? t0 : 0.f;
    t1 = ok ? t1 : 0.f;
    t2 = ok ? t2 : 0.f;
    t3 = ok ? t3 : 0.f;
  }
  v2f xs, xp;
  xs.x = t0;
  xs.y = lanelow ? t1 : 0.f;  // K=3 padded to 4
  xp.x = t2;
  xp.y = lanelow ? t3 : 0.f;

  // spatial branch
  v8f a1 = relu8(wmma4(xs, F.sw1f, splat8(F.sb1v)));
  store_ctile(tile, a1, lane);
  v8f s2 = splat8(F.sb2v);
#pragma unroll
  for (int kk = 0; kk < 3; ++kk)
    s2 = wmma4(load_afrag(tile, TILE_STRIDE, 4 * kk, lane), F.sw2f[kk], s2);
  // param branch (reuses tile; per-wave DS ops are in-order)
  v8f b1 = relu8(wmma4(xp, F.pw1f, splat8(F.pb1v)));
  store_ctile(tile, b1, lane);
  v8f p2 = splat8(F.pb2v);
#pragma unroll
  for (int kk = 0; kk < 3; ++kk)
    p2 = wmma4(load_afrag(tile, TILE_STRIDE, 4 * kk, lane), F.pw2f[kk], p2);
  sOut = s2;
  pOut = p2;
}

__global__ void init_norms_kernel(float* __restrict__ norms) {
  if (threadIdx.x < 2) norms[threadIdx.x] = 0.f;
}

__global__ void __launch_bounds__(256) pass1_kernel(
    const float* __restrict__ x, const float* __restrict__ sw1,
    const float* __restrict__ sb1, const float* __restrict__ sw2,
    const float* __restrict__ sb2, const float* __restrict__ pw1,
    const float* __restrict__ pb1, const float* __restrict__ pw2,
    const float* __restrict__ pb2, float* __restrict__ norms, int Nrows) {
  __shared__ float lds[8][16 * TILE_STRIDE];
  int lane = threadIdx.x & 31;
  int wave = threadIdx.x >> 5;
  float* tile = lds[wave];

  SPFrags F;
  load_sp_frags(F, sw1, sb1, sw2, sb2, pw1, pb1, pw2, pb2, lane);

  int nt  = (Nrows + 15) >> 4;
  int wid = blockIdx.x * 8 + wave;
  int nw  = gridDim.x * 8;
  int mb  = (lane >> 4) * 8;
  int kq  = 2 * (lane >> 4);        // 0 or 2
  int c0  = kq;
  int c1  = (kq + 1 < 3) ? kq + 1 : 2;
  bool lanelow = (lane < 16);

  float ssum = 0.f, psum = 0.f;
  for (int t = wid; t < nt; t += nw) {
    long r0  = (long)t << 4;
    int rows = Nrows - (int)r0;
    // prefetch next tile's x rows (one row per lane; lanes 0-15 cover the 384B tile)
    if (t + nw < nt)
      __builtin_prefetch(x + (((long)(t + nw) << 4) + lane) * 6, 0, 0);
    v8f s, p;
    if (rows >= 16) {
      compute_sp<true>(x, r0, 16, F, tile, lane, c0, c1, lanelow, s, p);
      // padding columns are exactly 0 -> no column mask needed
#pragma unroll
      for (int v = 0; v < 8; ++v) {
        ssum = fmaf(s[v], s[v], ssum);
        psum = fmaf(p[v], p[v], psum);
      }
    } else {
      compute_sp<false>(x, r0, rows, F, tile, lane, c0, c1, lanelow, s, p);
#pragma unroll
      for (int v = 0; v < 8; ++v) {
        float rmask = (mb + v < rows) ? 1.f : 0.f;
        ssum = fmaf(s[v] * rmask, s[v], ssum);
        psum = fmaf(p[v] * rmask, p[v], psum);
      }
    }
  }
#pragma unroll
  for (int off = 16; off >= 1; off >>= 1) {
    ssum += __shfl_xor(ssum, off, 32);
    psum += __shfl_xor(psum, off, 32);
  }
  if (lane == 0) {
    atomicAdd(&norms[0], ssum);
    atomicAdd(&norms[1], psum);
  }
}

__global__ void __launch_bounds__(128) pass2_kernel(
    const float* __restrict__ x, const float* __restrict__ sw1,
    const float* __restrict__ sb1, const float* __restrict__ sw2,
    const float* __restrict__ sb2, const float* __restrict__ pw1,
    const float* __restrict__ pb1, const float* __restrict__ pw2,
    const float* __restrict__ pb2, const float* __restrict__ ow1,
    const float* __restrict__ ob1, const float* __restrict__ ow2,
    const float* __restrict__ ob2, const float* __restrict__ norms,
    float* __restrict__ out, int Nrows) {
  __shared__ float lds_t[4][16 * TILE_STRIDE];
  __shared__ float lds_h[4][16 * H_STRIDE];
  int lane = threadIdx.x & 31;
  int wave = threadIdx.x >> 5;
  float* tile = lds_t[wave];
  float* hbuf = lds_h[wave];

  SPFrags F;
  load_sp_frags(F, sw1, sb1, sw2, sb2, pw1, pb1, pw2, pb2, lane);

  // ow1: 10x100 row-major -> 7 N-tiles x 3 K-steps; ow2: 100x3 -> 25 K-steps.
  v2f ow1f[7][3];
  float ob1v[7];
#pragma unroll
  for (int j = 0; j < 7; ++j) {
    ob1v[j] = load_biaslane(ob1, j * 16, 100, lane);
#pragma unroll
    for (int kk = 0; kk < 3; ++kk)
      ow1f[j][kk] = load_bfrag(ow1, 100, 4 * kk, j * 16, 10, 100, lane);
  }
  v2f ow2f[25];
#pragma unroll
  for (int kk = 0; kk < 25; ++kk)
    ow2f[kk] = load_bfrag(ow2, 3, 4 * kk, 0, 100, 3, lane);
  float ob2v = load_biaslane(ob2, 0, 3, lane);

  // (s/|s|)*(p/|p|) = s*p*rsqrt(sum_s2 * sum_p2)
  float scale = rsqrtf(norms[0] * norms[1]);

  int nt  = (Nrows + 15) >> 4;
  int wid = blockIdx.x * 4 + wave;
  int nw  = gridDim.x * 4;
  int n   = lane & 15;
  int mb  = (lane >> 4) * 8;
  int kq  = 2 * (lane >> 4);
  int c0  = kq;
  int c1  = (kq + 1 < 3) ? kq + 1 : 2;
  bool lanelow = (lane < 16);

  for (int t = wid; t < nt; t += nw) {
    long r0  = (long)t << 4;
    int rows = Nrows - (int)r0;
    bool full = (rows >= 16);
    if (!full && rows < 16) rows = rows;  // keep rows as-is
    if (t + nw < nt)
      __builtin_prefetch(x + (((long)(t + nw) << 4) + lane) * 6, 0, 0);
    v8f s, p;
    if (full)
      compute_sp<true>(x, r0, 16, F, tile, lane, c0, c1, lanelow, s, p);
    else
      compute_sp<false>(x, r0, rows, F, tile, lane, c0, c1, lanelow, s, p);

    v8f tf;
#pragma unroll
    for (int v = 0; v < 8; ++v) tf[v] = s[v] * p[v] * scale;
    store_ctile(tile, tf, lane);
    v2f ta[3];
#pragma unroll
    for (int kk = 0; kk < 3; ++kk)
      ta[kk] = load_afrag(tile, TILE_STRIDE, 4 * kk, lane);

    // hidden layer: 10 -> 100 (7 column tiles); cols >= 100 compute exactly 0.
#pragma unroll
    for (int j = 0; j < 7; ++j) {
      v8f h = splat8(ob1v[j]);
#pragma unroll
      for (int kk = 0; kk < 3; ++kk) h = wmma4(ta[kk], ow1f[j][kk], h);
      h = relu8(h);
#pragma unroll
      for (int v = 0; v < 8; ++v)
        hbuf[(mb + v) * H_STRIDE + j * 16 + n] = h[v];
    }

    // output layer: 100 -> 3 (K = 100 exactly, 25 K-steps)
    v8f o = splat8(ob2v);
#pragma unroll
    for (int kk = 0; kk < 25; ++kk)
      o = wmma4(load_afrag(hbuf, H_STRIDE, 4 * kk, lane), ow2f[kk], o);

    if (full) {
      if (n < 3) {
        float* ob = out + r0 * 3 + n;
#pragma unroll
        for (int v = 0; v < 8; ++v) ob[(mb + v) * 3] = o[v];
      }
    } else {
#pragma unroll
      for (int v = 0; v < 8; ++v) {
        int m = mb + v;
        if (n < 3 && m < rows) out[(r0 + m) * 3 + n] = o[v];
      }
    }
  }
}

extern "C" void kernel_launch(void* const* d_in, const int* in_sizes, int n_in,
                              void* d_out, int out_size, void* d_ws,
                              size_t ws_size, hipStream_t stream) {
  (void)n_in; (void)out_size; (void)ws_size;
  const float* x   = (const float*)d_in[0];
  const float* sw1 = (const float*)d_in[1];
  const float* sb1 = (const float*)d_in[2];
  const float* sw2 = (const float*)d_in[3];
  const float* sb2 = (const float*)d_in[4];
  const float* pw1 = (const float*)d_in[5];
  const float* pb1 = (const float*)d_in[6];
  const float* pw2 = (const float*)d_in[7];
  const float* pb2 = (const float*)d_in[8];
  const float* ow1 = (const float*)d_in[9];
  const float* ob1 = (const float*)d_in[10];
  const float* ow2 = (const float*)d_in[11];
  const float* ob2 = (const float*)d_in[12];
  float* out   = (float*)d_out;
  float* norms = (float*)d_ws;  // 2 floats of scratch

  int Nrows = in_sizes[0] / 6;
  int nt    = (Nrows + 15) / 16;

  init_norms_kernel<<<1, 32, 0, stream>>>(norms);

  int b1 = (nt + 7) / 8;
  if (b1 > 1024) b1 = 1024;
  if (b1 < 1) b1 = 1;
  pass1_kernel<<<b1, 256, 0, stream>>>(x, sw1, sb1, sw2, sb2, pw1, pb1, pw2,
                                       pb2, norms, Nrows);

  int b2 = (nt + 3) / 4;
  if (b2 > 2048) b2 = 2048;
  if (b2 < 1) b2 = 1;
  pass2_kernel<<<b2, 128, 0, stream>>>(x, sw1, sb1, sw2, sb2, pw1, pb1, pw2,
                                       pb2, ow1, ob1, ow2, ob2, norms, out,
                                       Nrows);
}